// TimeDomainModel_43662637531488
// MI455X (gfx1250) — compile-verified
//
#include <hip/hip_runtime.h>
#include <hip/hip_bf16.h>

// Balloon-Windkessel BOLD model, 1000 Euler steps, single scalar param (mtt).
// Strictly serial recurrence -> one wave32, state lives in registers.
// Observation y = w . [1, q2, q2/v2, v2] is affine, so we batch 16 steps and
// project them with one V_WMMA_F32_16X16X4_F32 (full-fp32 WMMA, K=4 == state
// width) per tile. Row 0 of D holds the 16 outputs (lanes 0-15, d[0]).
// Latency tuning: all per-step divisions replaced with native v_rcp_f32
// (single TRANS op) to shorten the serial dependency chain.

typedef __attribute__((ext_vector_type(2))) float v2f;
typedef __attribute__((ext_vector_type(8))) float v8f;

#define C_DT    0.01f
#define C_U     1.0f
#define C_KAPPA 0.65f
#define C_GAMMA 0.41f
#define C_RHO   0.34f
#define C_V0    0.02f
// inv_alpha = 1/0.32 = 3.125 exactly
#define C_INV_ALPHA 3.125f
// ln(1 - RHO) = ln(0.66)
#define C_LN_1MRHO (-0.4155154439616658f)

__global__ __launch_bounds__(32)
void bold_euler_wmma_kernel(const float* __restrict__ mtt_p,
                            float* __restrict__ out, int n) {
    const int lane = threadIdx.x;        // wave32: one full wave, EXEC all-1s

    const float mtt     = mtt_p[0];
    const float inv_mtt = __builtin_amdgcn_rcpf(mtt);
    const float inv_rho = 1.0f / C_RHO;  // compile-time constant fold

    // Observation weights: y = W0 + W1*q2 + W2*(q2/v2) + W3*v2
    const float k1 = 7.0f * C_RHO;
    const float k2 = 2.0f;
    const float k3 = 2.0f * C_RHO - 0.2f;
    const float W0 =  C_V0 * (k1 + k2 + k3);
    const float W1 = -C_V0 * k1;
    const float W2 = -C_V0 * k2;
    const float W3 = -C_V0 * k3;

    // A fragment: 16x4 f32, every row = w.
    // Lanes 0-15: VGPR0=K0, VGPR1=K1.  Lanes 16-31: VGPR0=K2, VGPR1=K3.
    v2f a;
    a.x = (lane < 16) ? W0 : W2;
    a.y = (lane < 16) ? W1 : W3;

    // ODE state (identical in every lane; redundant compute keeps EXEC full)
    float s = 0.0f, f = 1.0f, v = 1.0f, q = 1.0f;

    const int ntiles = (n + 15) >> 4;    // 63 tiles for n=1000
    for (int t = 0; t < ntiles; ++t) {
        // B fragment: 4x16 f32 column j = [1, q2_j, r_j, v2_j]
        // Lanes 0-15 (col=lane): VGPR0=K0, VGPR1=K1
        // Lanes 16-31 (col=lane-16): VGPR0=K2, VGPR1=K3
        v2f b;
        b.x = 0.0f;
        b.y = 0.0f;

        #pragma unroll
        for (int j = 0; j < 16; ++j) {
            // derivatives from pre-update state (single-TRANS reciprocals)
            float inv_v = __builtin_amdgcn_rcpf(v);
            float inv_f = __builtin_amdgcn_rcpf(f);
            float fv = __expf(C_INV_ALPHA * __logf(v));   // v^(1/alpha)
            float E  = 1.0f - __expf(C_LN_1MRHO * inv_f); // 1-(1-rho)^(1/f)
            float ds = C_U - C_KAPPA * s - C_GAMMA * (f - 1.0f);
            float df = s;
            float dv = (f - fv) * inv_mtt;
            float dq = (f * E * inv_rho - q * fv * inv_v) * inv_mtt;
            // simultaneous Euler update
            s += C_DT * ds;
            f += C_DT * df;
            v += C_DT * dv;
            q += C_DT * dq;

            float r = q * __builtin_amdgcn_rcpf(v);       // q2 / v2
            if (lane == j)      { b.x = 1.0f; b.y = q; }  // K0, K1
            if (lane == j + 16) { b.x = r;    b.y = v; }  // K2, K3
        }

        // D = A x B + 0 : row 0 of D = 16 outputs of this tile.
        v8f c = {};
        c = __builtin_amdgcn_wmma_f32_16x16x4_f32(
                /*neg_a=*/false, a, /*neg_b=*/false, b,
                /*c_mod=*/(short)0, c, /*reuse_a=*/false, /*reuse_b=*/false);

        // C/D layout: VGPR0, lanes 0-15 => M=0, N=lane.
        int idx = (t << 4) + lane;
        if (lane < 16 && idx < n) out[idx] = c[0];
    }
}

extern "C" void kernel_launch(void* const* d_in, const int* in_sizes, int n_in,
                              void* d_out, int out_size, void* d_ws, size_t ws_size,
                              hipStream_t stream) {
    (void)in_sizes; (void)n_in; (void)d_ws; (void)ws_size;
    const float* mtt = (const float*)d_in[0];
    float* out = (float*)d_out;
    bold_euler_wmma_kernel<<<1, 32, 0, stream>>>(mtt, out, out_size);
}